// Compatibility_11527692222836
// MI455X (gfx1250) — compile-verified
//
#include <hip/hip_runtime.h>

typedef __attribute__((ext_vector_type(2))) float v2f;
typedef __attribute__((ext_vector_type(8))) float v8f;

#define N_PTS   2048
#define DIMS    16
#define NTILE   (N_PTS / 16)      // 128 tiles per dimension
#define NCHUNK  32                // n-tiles handled per wave (gridDim.y = 4)
#define A_LO    (-9.0f)
#define A_SPAN  (18.0f)

// ---------------------------------------------------------------------------
// Helpers shared by both passes: each wave owns one 16-row Q strip.
// WMMA f32 16x16x4 layouts (ISA 7.12.2):
//   A (16x4):  lane L, row M=L&15, VGPR{0,1} = K = (L>>4)*2 + {0,1}
//   B (4x16):  lane L, col N=L&15, VGPR{0,1} = K = (L>>4)*2 + {0,1}
//   C (16x16): VGPR v, lane L: M = v + (L>>4)*8, N = L&15
// Since B = K^T, both A and B per-lane loads are 2 consecutive floats of a
// source row -> four float2 loads cover K=16 (4 chained WMMAs).
// ---------------------------------------------------------------------------

__device__ __forceinline__ float rowSq8(v2f x0, v2f x1, v2f x2, v2f x3) {
    return x0.x * x0.x + x0.y * x0.y + x1.x * x1.x + x1.y * x1.y +
           x2.x * x2.x + x2.y * x2.y + x3.x * x3.x + x3.y * x3.y;
}

__device__ __forceinline__ v8f tile_cross(v2f a0, v2f a1, v2f a2, v2f a3,
                                          v2f b0, v2f b1, v2f b2, v2f b3) {
    v8f c = {0.f, 0.f, 0.f, 0.f, 0.f, 0.f, 0.f, 0.f};
    c = __builtin_amdgcn_wmma_f32_16x16x4_f32(false, a0, false, b0, (short)0, c, false, false);
    c = __builtin_amdgcn_wmma_f32_16x16x4_f32(false, a1, false, b1, (short)0, c, false, false);
    c = __builtin_amdgcn_wmma_f32_16x16x4_f32(false, a2, false, b2, (short)0, c, false, false);
    c = __builtin_amdgcn_wmma_f32_16x16x4_f32(false, a3, false, b3, (short)0, c, false, false);
    return c;
}

// ---------------------------------------------------------------------------
// Init: per-slice min (bit pattern of +huge) / max (0) in workspace.
// ---------------------------------------------------------------------------
__global__ void compat_init_ws(unsigned int* __restrict__ wsMin,
                               unsigned int* __restrict__ wsMax) {
    int i = threadIdx.x;
    if (i < 32) {
        wsMin[i] = 0x7F7FFFFFu;   // FLT_MAX bit pattern
        wsMax[i] = 0u;            // 0.0f (distances are >= 0)
    }
}

// ---------------------------------------------------------------------------
// Pass 1: per-slice min/max of s = q^2 + k^2 - 2*cross (sqrt is monotone).
// ---------------------------------------------------------------------------
__global__ __launch_bounds__(256) void compat_minmax(const float* __restrict__ Q,
                                                     const float* __restrict__ K,
                                                     unsigned int* __restrict__ wsMin,
                                                     unsigned int* __restrict__ wsMax) {
    const int slice = blockIdx.z;
    const float* __restrict__ Qs = Q + (size_t)slice * N_PTS * DIMS;
    const float* __restrict__ Ks = K + (size_t)slice * N_PTS * DIMS;

    const int lane = threadIdx.x & 31;
    const int wave = threadIdx.x >> 5;
    const int r    = lane & 15;
    const int hi   = lane >> 4;         // 0 | 1
    const int koff = hi * 2;

    const int qbase = (blockIdx.x * 8 + wave) * 16;
    const float* qrow = Qs + (qbase + r) * DIMS + koff;
    v2f a0 = *(const v2f*)(qrow + 0);
    v2f a1 = *(const v2f*)(qrow + 4);
    v2f a2 = *(const v2f*)(qrow + 8);
    v2f a3 = *(const v2f*)(qrow + 12);

    float qpart = rowSq8(a0, a1, a2, a3);
    float qsq   = qpart + __shfl_xor(qpart, 16, 32);
    float qsq_m[8];
#pragma unroll
    for (int v = 0; v < 8; ++v)
        qsq_m[v] = __shfl(qsq, v + hi * 8, 32);

    float smin =  3.0e38f;
    float smax = -3.0e38f;

    const int ntile0 = blockIdx.y * NCHUNK;
    for (int t = 0; t < NCHUNK; ++t) {
        const int kbase = (ntile0 + t) * 16;
        const float* krow = Ks + (kbase + r) * DIMS + koff;
        v2f b0 = *(const v2f*)(krow + 0);
        v2f b1 = *(const v2f*)(krow + 4);
        v2f b2 = *(const v2f*)(krow + 8);
        v2f b3 = *(const v2f*)(krow + 12);

        float kpart = rowSq8(b0, b1, b2, b3);
        float ksq   = kpart + __shfl_xor(kpart, 16, 32);

        v8f c = tile_cross(a0, a1, a2, a3, b0, b1, b2, b3);
#pragma unroll
        for (int v = 0; v < 8; ++v) {
            float s = qsq_m[v] + ksq - 2.0f * c[v];
            smin = fminf(smin, s);
            smax = fmaxf(smax, s);
        }
    }

    // wave reduction
#pragma unroll
    for (int off = 16; off >= 1; off >>= 1) {
        smin = fminf(smin, __shfl_xor(smin, off, 32));
        smax = fmaxf(smax, __shfl_xor(smax, off, 32));
    }

    __shared__ float sMin[8], sMax[8];
    if (lane == 0) { sMin[wave] = smin; sMax[wave] = smax; }
    __syncthreads();

    if (threadIdx.x == 0) {
        float bmin = sMin[0], bmax = sMax[0];
#pragma unroll
        for (int w = 1; w < 8; ++w) {
            bmin = fminf(bmin, sMin[w]);
            bmax = fmaxf(bmax, sMax[w]);
        }
        float dmin = sqrtf(fmaxf(bmin, 0.0f));
        float dmax = sqrtf(fmaxf(bmax, 0.0f));
        // distances >= 0 -> IEEE bit pattern order == unsigned integer order
        atomicMin(&wsMin[slice], __float_as_uint(dmin));
        atomicMax(&wsMax[slice], __float_as_uint(dmax));
    }
}

// ---------------------------------------------------------------------------
// Pass 2: recompute tiles and stream the rescaled 512MB output (NT stores).
// ---------------------------------------------------------------------------
__global__ __launch_bounds__(256) void compat_write(const float* __restrict__ Q,
                                                    const float* __restrict__ K,
                                                    const unsigned int* __restrict__ wsMin,
                                                    const unsigned int* __restrict__ wsMax,
                                                    float* __restrict__ Out) {
    const int slice = blockIdx.z;
    const float* __restrict__ Qs = Q + (size_t)slice * N_PTS * DIMS;
    const float* __restrict__ Ks = K + (size_t)slice * N_PTS * DIMS;
    float* __restrict__ Os       = Out + (size_t)slice * N_PTS * N_PTS;

    const float minv  = __uint_as_float(wsMin[slice]);
    const float maxv  = __uint_as_float(wsMax[slice]);
    const float scale = A_SPAN / (maxv - minv);

    const int lane = threadIdx.x & 31;
    const int wave = threadIdx.x >> 5;
    const int r    = lane & 15;
    const int hi   = lane >> 4;
    const int koff = hi * 2;

    const int qbase = (blockIdx.x * 8 + wave) * 16;
    const float* qrow = Qs + (qbase + r) * DIMS + koff;
    v2f a0 = *(const v2f*)(qrow + 0);
    v2f a1 = *(const v2f*)(qrow + 4);
    v2f a2 = *(const v2f*)(qrow + 8);
    v2f a3 = *(const v2f*)(qrow + 12);

    float qpart = rowSq8(a0, a1, a2, a3);
    float qsq   = qpart + __shfl_xor(qpart, 16, 32);
    float qsq_m[8];
#pragma unroll
    for (int v = 0; v < 8; ++v)
        qsq_m[v] = __shfl(qsq, v + hi * 8, 32);

    const int ntile0 = blockIdx.y * NCHUNK;
    for (int t = 0; t < NCHUNK; ++t) {
        const int kbase = (ntile0 + t) * 16;
        const float* krow = Ks + (kbase + r) * DIMS + koff;
        v2f b0 = *(const v2f*)(krow + 0);
        v2f b1 = *(const v2f*)(krow + 4);
        v2f b2 = *(const v2f*)(krow + 8);
        v2f b3 = *(const v2f*)(krow + 12);

        float kpart = rowSq8(b0, b1, b2, b3);
        float ksq   = kpart + __shfl_xor(kpart, 16, 32);

        v8f c = tile_cross(a0, a1, a2, a3, b0, b1, b2, b3);

        float* orow = Os + (size_t)(qbase + hi * 8) * N_PTS + kbase + r;
#pragma unroll
        for (int v = 0; v < 8; ++v) {
            float s = qsq_m[v] + ksq - 2.0f * c[v];
            float d = sqrtf(fmaxf(s, 0.0f));
            float o = A_LO + (d - minv) * scale;
            __builtin_nontemporal_store(o, orow + (size_t)v * N_PTS);
        }
    }
}

// ---------------------------------------------------------------------------
extern "C" void kernel_launch(void* const* d_in, const int* in_sizes, int n_in,
                              void* d_out, int out_size, void* d_ws, size_t ws_size,
                              hipStream_t stream) {
    (void)in_sizes; (void)n_in; (void)out_size; (void)ws_size;
    const float* Q = (const float*)d_in[0];
    const float* K = (const float*)d_in[1];
    float* Out = (float*)d_out;

    unsigned int* wsMin = (unsigned int*)d_ws;       // 32 words
    unsigned int* wsMax = wsMin + 32;                // 32 words

    compat_init_ws<<<1, 32, 0, stream>>>(wsMin, wsMax);

    dim3 grid(NTILE / 8, NTILE / NCHUNK, 32);        // (16, 4, 32)
    dim3 block(256);
    compat_minmax<<<grid, block, 0, stream>>>(Q, K, wsMin, wsMax);
    compat_write <<<grid, block, 0, stream>>>(Q, K, wsMin, wsMax, Out);
}